// Grad_36361193128237
// MI455X (gfx1250) — compile-verified
//
#include <hip/hip_runtime.h>
#include <stdint.h>

// Problem geometry (fixed by the reference): (N=2, C=1, D=128, H=256, W=256) f32.
#define N_BATCH 2
#define D_DIM   128
#define H_DIM   256
#define W_DIM   256
#define TILE    16
#define ZCHUNK  32
#define NZCH    (D_DIM / ZCHUNK)        // 4 z-chunks
#define PATCH   18                      // 16 + halo
#define PLANE_ELEMS (PATCH * PATCH)     // 324
#define NSLOT   4                       // raw z-plane ring (enables async overlap)
#define NIMG    3
#define IMG_STRIDE (NSLOT * PLANE_ELEMS)          // 1296 floats per image region
#define SMEM_ELEMS (NIMG * IMG_STRIDE)            // 3888 floats = 15552 B

typedef float v2f __attribute__((ext_vector_type(2)));
typedef float v8f __attribute__((ext_vector_type(8)));

// ---- CDNA5 async global->LDS copy (ASYNCcnt-tracked), GV addressing mode ----
__device__ __forceinline__ void async_ld_f32(unsigned lds_addr, const float* g) {
  asm volatile("global_load_async_to_lds_b32 %0, %1, off"
               :: "v"(lds_addr), "v"((unsigned long long)(uintptr_t)g)
               : "memory");
}
__device__ __forceinline__ void wait_async0() {
  asm volatile("s_wait_asynccnt 0" ::: "memory");
}

// ---- Wave reduction through the matrix pipe: V_WMMA_F32_16X16X4_F32 ----
// A: lane L contributes {v, 0} -> each lane's value appears exactly once in A.
// B: all-ones 4x16 (layout independent). D[m][n] = s_m + s_{m+16} (n-invariant).
// Lane L holds D rows M = 8*(L/16)..8*(L/16)+7 -> summing its 8 regs gives a
// half-sum; one shfl_xor(16) merges the halves. Exact f32 arithmetic.
__device__ __forceinline__ float wave_sum_wmma(float v) {
  v2f a; a[0] = v;    a[1] = 0.0f;
  v2f b; b[0] = 1.0f; b[1] = 1.0f;
  v8f c = {};
  v8f d = __builtin_amdgcn_wmma_f32_16x16x4_f32(
      /*neg_a=*/false, a, /*neg_b=*/false, b,
      /*c_mod=*/(short)0, c, /*reuse_a=*/false, /*reuse_b=*/false);
  float p = ((d[0] + d[1]) + (d[2] + d[3])) + ((d[4] + d[5]) + (d[6] + d[7]));
  p += __shfl_xor(p, 16, 32);
  return p;  // all lanes hold the wave total
}

// Per-plane separable pieces from the 3x3 raw neighborhood in LDS:
//   a = SxSy, b = DxSy (x-derivative), c = SxDy (y-derivative)
struct ABC { float a, b, c; };

__device__ __forceinline__ ABC plane_abc(const float* P, int corner) {
  const float* r0 = P + corner;        // top-left of the 3x3 window
  float p00 = r0[0],  p01 = r0[1],  p02 = r0[2];
  float p10 = r0[18], p11 = r0[19], p12 = r0[20];
  float p20 = r0[36], p21 = r0[37], p22 = r0[38];
  float s0 = p00 + 2.0f * p01 + p02;
  float s1 = p10 + 2.0f * p11 + p12;
  float s2 = p20 + 2.0f * p21 + p22;
  float d0 = p00 - p02, d1 = p10 - p12, d2 = p20 - p22;
  ABC r;
  r.a = s0 + 2.0f * s1 + s2;   // SxSy
  r.b = d0 + 2.0f * d1 + d2;   // DxSy
  r.c = s0 - s2;               // SxDy
  return r;
}

// Combine a 3-plane register history into |d/dx|+|d/dy|+|d/dz|.
__device__ __forceinline__ float combine(const ABC& m, const ABC& c0, const ABC& p) {
  float gx = m.b + 2.0f * c0.b + p.b;   // Dx Sy Sz
  float gy = m.c + 2.0f * c0.c + p.c;   // Sx Dy Sz
  float gz = p.a - m.a;                 // Sx Sy Dz
  return fabsf(gx) + fabsf(gy) + fabsf(gz);
}

__global__ __launch_bounds__(256)
void sobel_fuse_kernel(const float* __restrict__ A, const float* __restrict__ B,
                       const float* __restrict__ F, float* __restrict__ ws) {
  __shared__ float smem[SMEM_ELEMS];
  __shared__ float red[8];

  const int tid = threadIdx.x;
  const int xt  = blockIdx.x * TILE;
  const int yt  = blockIdx.y * TILE;
  const int n   = blockIdx.z / NZCH;
  const int z0  = (blockIdx.z % NZCH) * ZCHUNK;
  const float* img0 = A; const float* img1 = B; const float* img2 = F;

  // Zero LDS once: covers the fixed out-of-range halo cells (never overwritten)
  // and the virtual z=-1 plane; barrier commits the ds stores before any async
  // DMA can target the same addresses.
  for (int j = tid; j < SMEM_ELEMS; j += 256) smem[j] = 0.0f;
  __syncthreads();

  auto stage = [&](int z) {
    const int slot = z & 3;
    if (z >= 0 && z < D_DIM) {
      for (int e = tid; e < PLANE_ELEMS; e += 256) {
        int r = e / PATCH, c2 = e - r * PATCH;
        int gy = yt + r - 1, gx = xt + c2 - 1;
        if ((unsigned)gy < (unsigned)H_DIM && (unsigned)gx < (unsigned)W_DIM) {
          size_t go = ((size_t)((n * D_DIM + z) * H_DIM + gy)) * (size_t)W_DIM + gx;
          unsigned l0 = (unsigned)(uintptr_t)(&smem[0 * IMG_STRIDE + slot * PLANE_ELEMS + e]);
          unsigned l1 = (unsigned)(uintptr_t)(&smem[1 * IMG_STRIDE + slot * PLANE_ELEMS + e]);
          unsigned l2 = (unsigned)(uintptr_t)(&smem[2 * IMG_STRIDE + slot * PLANE_ELEMS + e]);
          async_ld_f32(l0, img0 + go);
          async_ld_f32(l1, img1 + go);
          async_ld_f32(l2, img2 + go);
        }
      }
    } else {
      // z out of volume: zero-padding plane (slot may be dirty mid-chunk)
      for (int e = tid; e < PLANE_ELEMS; e += 256) {
        smem[0 * IMG_STRIDE + slot * PLANE_ELEMS + e] = 0.0f;
        smem[1 * IMG_STRIDE + slot * PLANE_ELEMS + e] = 0.0f;
        smem[2 * IMG_STRIDE + slot * PLANE_ELEMS + e] = 0.0f;
      }
    }
  };

  // Prologue: raw planes z0-1, z0, z0+1 resident before the loop.
  stage(z0 - 1);
  stage(z0);
  stage(z0 + 1);
  wait_async0();
  __syncthreads();

  const int y = tid >> 4;
  const int x = tid & 15;
  const int corner = y * PATCH + x;    // top-left of this thread's 3x3 window
  float acc = 0.0f;

  // Register ring of xy-filtered triples for planes z-1 and z (per image).
  ABC Am, A0, Bm, B0, Fm, F0;
  {
    const int smi = (z0 - 1) & 3, s0i = z0 & 3;
    Am = plane_abc(smem + 0 * IMG_STRIDE + smi * PLANE_ELEMS, corner);
    Bm = plane_abc(smem + 1 * IMG_STRIDE + smi * PLANE_ELEMS, corner);
    Fm = plane_abc(smem + 2 * IMG_STRIDE + smi * PLANE_ELEMS, corner);
    A0 = plane_abc(smem + 0 * IMG_STRIDE + s0i * PLANE_ELEMS, corner);
    B0 = plane_abc(smem + 1 * IMG_STRIDE + s0i * PLANE_ELEMS, corner);
    F0 = plane_abc(smem + 2 * IMG_STRIDE + s0i * PLANE_ELEMS, corner);
  }

  for (int zc = z0; zc < z0 + ZCHUNK; ++zc) {
    // Kick the async DMA for raw plane zc+2 (ring slot not read this
    // iteration), overlapping it with the stencil compute below.
    if (zc + 2 <= z0 + ZCHUNK) stage(zc + 2);

    // Only the NEW plane zc+1 needs LDS reads: 9 per image.
    const int spi = (zc + 1) & 3;
    ABC Ap = plane_abc(smem + 0 * IMG_STRIDE + spi * PLANE_ELEMS, corner);
    ABC Bp = plane_abc(smem + 1 * IMG_STRIDE + spi * PLANE_ELEMS, corner);
    ABC Fp = plane_abc(smem + 2 * IMG_STRIDE + spi * PLANE_ELEMS, corner);

    float ga = combine(Am, A0, Ap);
    float gb = combine(Bm, B0, Bp);
    float gf = combine(Fm, F0, Fp);
    acc += fabsf(gf - fmaxf(ga, gb));

    // Advance the register ring.
    Am = A0; A0 = Ap;
    Bm = B0; B0 = Bp;
    Fm = F0; F0 = Fp;

    wait_async0();
    __syncthreads();
  }

  // Deterministic block reduction: WMMA wave-sum -> 8 wave totals -> thread 0.
  float wsum = wave_sum_wmma(acc);
  if ((tid & 31) == 0) red[tid >> 5] = wsum;
  __syncthreads();
  if (tid == 0) {
    float s = 0.0f;
#pragma unroll
    for (int i = 0; i < 8; ++i) s += red[i];
    ws[(blockIdx.z * gridDim.y + blockIdx.y) * gridDim.x + blockIdx.x] = s;
  }
}

__global__ __launch_bounds__(256)
void finalize_kernel(const float* __restrict__ ws, float* __restrict__ out,
                     int nblocks, float scale) {
  __shared__ float red[8];
  float acc = 0.0f;
  for (int i = threadIdx.x; i < nblocks; i += 256) acc += ws[i];
  float wsum = wave_sum_wmma(acc);
  if ((threadIdx.x & 31) == 0) red[threadIdx.x >> 5] = wsum;
  __syncthreads();
  if (threadIdx.x == 0) {
    float s = 0.0f;
#pragma unroll
    for (int i = 0; i < 8; ++i) s += red[i];
    out[0] = s * scale;  // overwrite (d_out is poisoned), never accumulate
  }
}

extern "C" void kernel_launch(void* const* d_in, const int* in_sizes, int n_in,
                              void* d_out, int out_size, void* d_ws, size_t ws_size,
                              hipStream_t stream) {
  (void)in_sizes; (void)n_in; (void)out_size; (void)ws_size;
  const float* A = (const float*)d_in[0];
  const float* B = (const float*)d_in[1];
  const float* F = (const float*)d_in[2];
  float* ws  = (float*)d_ws;
  float* out = (float*)d_out;

  dim3 grid(W_DIM / TILE, H_DIM / TILE, N_BATCH * NZCH);  // 16 x 16 x 8 = 2048 blocks
  sobel_fuse_kernel<<<grid, 256, 0, stream>>>(A, B, F, ws);

  const int nblocks = grid.x * grid.y * grid.z;
  const float scale = 1.0f / (float)((size_t)N_BATCH * D_DIM * H_DIM * W_DIM);
  finalize_kernel<<<1, 256, 0, stream>>>(ws, out, nblocks, scale);
}